// GAT_62268435857750
// MI455X (gfx1250) — compile-verified
//
#include <hip/hip_runtime.h>
#include <math.h>

// ---------------------------------------------------------------------------
// GAT 2-layer forward for gfx1250 (CDNA5, wave32).
// GEMMs use V_WMMA_F32_16X16X4_F32 (f32 inputs -> full precision, and the
// GEMM is bandwidth-bound anyway: X is 205MB @ 23.3TB/s ~ 9us).
// Edge softmax/aggregation: 3-pass atomic segment softmax with 128-bit
// vectorized gather in the aggregation pass.
// ---------------------------------------------------------------------------

typedef __attribute__((ext_vector_type(2))) float v2f;
typedef __attribute__((ext_vector_type(8))) float v8f;

#define F_IN   512
#define HID    8
#define HEADS  8
#define HC1    64      // HEADS*HID
#define NCLS   16
#define NEG_SLOPE 0.2f

// ---------------------------------------------------------------------------
// WMMA f32 GEMM, layer 1: H1[N,64] = X[N,512] @ W1[512,64]
// One wave computes a full 16x64 output tile with 4 WMMA accumulators so the
// A fragment (the big X operand) is loaded exactly once and reused across all
// 4 column tiles. Block = 256 threads = 8 waves = 8 row tiles.
//
// A (16x4 f32) layout: lanes 0-15 hold M=lane, K={0,1}; lanes 16-31 hold
// M=lane-16, K={2,3}.  B (4x16): lanes 0-15 rows K=0/1, lanes 16-31 rows
// K=2/3, N=lane&15.  C/D: VGPR v = row (v + 8*(lane>>4)), col = lane&15.
// ---------------------------------------------------------------------------
__global__ void gat_gemm1_wmma(const float* __restrict__ X,
                               const float* __restrict__ W,
                               float* __restrict__ H1, int nRowTiles) {
  const int wave = threadIdx.x >> 5;
  const int rt   = blockIdx.x * (blockDim.x >> 5) + wave;
  if (rt >= nRowTiles) return;            // wave-uniform guard: EXEC stays all-1s
  const int lane    = threadIdx.x & 31;
  const int halfsel = lane >> 4;          // 0: lanes 0-15, 1: lanes 16-31
  const int lrow    = lane & 15;
  const int klo     = halfsel * 2;
  const int row0    = rt * 16;

  const float* __restrict__ xrow = X + (size_t)(row0 + lrow) * F_IN;

  v8f acc0 = {}, acc1 = {}, acc2 = {}, acc3 = {};
  for (int kb = 0; kb < F_IN; kb += 64) {
    // Prefetch the next 64-float K-chunk of this lane's X row (speculative,
    // harmlessly dropped past the end).  Lowers to global_prefetch_b8.
    __builtin_prefetch(xrow + kb + 64, 0, 1);
#pragma unroll
    for (int kk = kb; kk < kb + 64; kk += 4) {
      v2f a;
      a.x = xrow[kk + klo];
      a.y = xrow[kk + klo + 1];
      const float* __restrict__ wr = W + (size_t)(kk + klo) * HC1 + lrow;
      v2f b0, b1, b2, b3;
      b0.x = wr[0];        b0.y = wr[HC1];
      b1.x = wr[16];       b1.y = wr[HC1 + 16];
      b2.x = wr[32];       b2.y = wr[HC1 + 32];
      b3.x = wr[48];       b3.y = wr[HC1 + 48];
      acc0 = __builtin_amdgcn_wmma_f32_16x16x4_f32(false, a, false, b0,
                                                   (short)0, acc0, false, false);
      acc1 = __builtin_amdgcn_wmma_f32_16x16x4_f32(false, a, false, b1,
                                                   (short)0, acc1, false, false);
      acc2 = __builtin_amdgcn_wmma_f32_16x16x4_f32(false, a, false, b2,
                                                   (short)0, acc2, false, false);
      acc3 = __builtin_amdgcn_wmma_f32_16x16x4_f32(false, a, false, b3,
                                                   (short)0, acc3, false, false);
    }
  }
#pragma unroll
  for (int v = 0; v < 8; ++v) {
    const int r = row0 + v + 8 * halfsel;
    float* __restrict__ o = H1 + (size_t)r * HC1 + lrow;
    o[0]  = acc0[v];
    o[16] = acc1[v];
    o[32] = acc2[v];
    o[48] = acc3[v];
  }
}

// ---------------------------------------------------------------------------
// WMMA f32 GEMM, layer 2: H2[N,16] = Hact[N,64] @ W2[64,16]
// Block = 256 threads = 8 waves; each wave does one 16-row tile (16 WMMAs).
// ---------------------------------------------------------------------------
__global__ void gat_gemm2_wmma(const float* __restrict__ Hin,
                               const float* __restrict__ W,
                               float* __restrict__ H2, int nRowTiles) {
  const int wave = threadIdx.x >> 5;
  const int rt   = blockIdx.x * (blockDim.x >> 5) + wave;
  if (rt >= nRowTiles) return;
  const int lane    = threadIdx.x & 31;
  const int halfsel = lane >> 4;
  const int lrow    = lane & 15;
  const int klo     = halfsel * 2;
  const int row0    = rt * 16;

  const float* __restrict__ hrow = Hin + (size_t)(row0 + lrow) * HC1;

  v8f acc = {};
#pragma unroll
  for (int kk = 0; kk < HC1; kk += 4) {
    v2f a, b;
    a.x = hrow[kk + klo];
    a.y = hrow[kk + klo + 1];
    b.x = W[(kk + klo) * NCLS + lrow];
    b.y = W[(kk + klo + 1) * NCLS + lrow];
    acc = __builtin_amdgcn_wmma_f32_16x16x4_f32(false, a, false, b,
                                                (short)0, acc, false, false);
  }
#pragma unroll
  for (int v = 0; v < 8; ++v) {
    const int r = row0 + v + 8 * halfsel;
    H2[(size_t)r * NCLS + lrow] = acc[v];
  }
}

// ---------------------------------------------------------------------------
// Per-node attention logit halves, layer 1: one thread per (node, head).
// ---------------------------------------------------------------------------
__global__ void gat_logits1(const float* __restrict__ H1,
                            const float* __restrict__ asrc,
                            const float* __restrict__ adst,
                            float* __restrict__ als, float* __restrict__ ald,
                            int N) {
  const int gid = blockIdx.x * blockDim.x + threadIdx.x;
  if (gid >= N * HEADS) return;
  const int n = gid / HEADS, h = gid % HEADS;
  const float* hp = H1 + (size_t)n * HC1 + h * HID;
  float s = 0.f, d = 0.f;
#pragma unroll
  for (int c = 0; c < HID; ++c) {
    s += hp[c] * asrc[h * HID + c];
    d += hp[c] * adst[h * HID + c];
  }
  als[gid] = s;
  ald[gid] = d;
}

// Layer 2 (1 head, 16 channels): one thread per node.
__global__ void gat_logits2(const float* __restrict__ H2,
                            const float* __restrict__ asrc,
                            const float* __restrict__ adst,
                            float* __restrict__ als, float* __restrict__ ald,
                            int N) {
  const int n = blockIdx.x * blockDim.x + threadIdx.x;
  if (n >= N) return;
  const float* hp = H2 + (size_t)n * NCLS;
  float s = 0.f, d = 0.f;
#pragma unroll
  for (int c = 0; c < NCLS; ++c) {
    s += hp[c] * asrc[c];
    d += hp[c] * adst[c];
  }
  als[n] = s;
  ald[n] = d;
}

// ---------------------------------------------------------------------------
// Fill helper (graph-capture safe; no hipMemset needed).
// ---------------------------------------------------------------------------
__global__ void fill_f32(float* __restrict__ p, float v, int n) {
  const int i = blockIdx.x * blockDim.x + threadIdx.x;
  if (i < n) p[i] = v;
}

// Float atomic max via signed/unsigned integer monotonic trick.
__device__ __forceinline__ void atomicMaxF(float* addr, float val) {
  if (val >= 0.f) atomicMax((int*)addr, __float_as_int(val));
  else            atomicMin((unsigned int*)addr, __float_as_uint(val));
}

__device__ __forceinline__ float leaky(float x) {
  return x > 0.f ? x : NEG_SLOPE * x;
}

// ---------------------------------------------------------------------------
// Segment softmax pass A: per-(edge, head) atomic max of leaky logits.
// Edges e in [0,E) come from edge_index; e in [E, E+N) are self loops.
// ---------------------------------------------------------------------------
__global__ void gat_edge_max(const int* __restrict__ ei,
                             const float* __restrict__ als,
                             const float* __restrict__ ald,
                             float* __restrict__ m, int E, int N, int H) {
  const int gid = blockIdx.x * blockDim.x + threadIdx.x;
  const int total = (E + N) * H;
  if (gid >= total) return;
  const int e = gid / H, h = gid % H;
  int src, dst;
  if (e < E) { src = ei[e]; dst = ei[E + e]; } else { src = dst = e - E; }
  const float x = leaky(als[src * H + h] + ald[dst * H + h]);
  atomicMaxF(&m[dst * H + h], x);
}

// Pass B: per-(edge, head) atomic add of exp(e - max).
__global__ void gat_edge_expsum(const int* __restrict__ ei,
                                const float* __restrict__ als,
                                const float* __restrict__ ald,
                                const float* __restrict__ m,
                                float* __restrict__ denom, int E, int N, int H) {
  const int gid = blockIdx.x * blockDim.x + threadIdx.x;
  const int total = (E + N) * H;
  if (gid >= total) return;
  const int e = gid / H, h = gid % H;
  int src, dst;
  if (e < E) { src = ei[e]; dst = ei[E + e]; } else { src = dst = e - E; }
  const float x = leaky(als[src * H + h] + ald[dst * H + h]);
  atomicAdd(&denom[dst * H + h], __expf(x - m[dst * H + h]));
}

// ---------------------------------------------------------------------------
// Pass C: weighted scatter-add of messages.  One thread per (edge, 4-channel
// quad): one global_load_b128 gather of the message slice, one exp per quad,
// 4 global_atomic_add_f32.  Requires C % 4 == 0 (C is 8 or 16 here).
// ---------------------------------------------------------------------------
__global__ void gat_edge_agg4(const int* __restrict__ ei,
                              const float* __restrict__ als,
                              const float* __restrict__ ald,
                              const float* __restrict__ m,
                              const float* __restrict__ denom,
                              const float* __restrict__ feat,
                              float* __restrict__ agg,
                              int E, int N, int H, int C) {
  const int HC = H * C;
  const int QP = HC >> 2;                    // quads per edge
  const int gid = blockIdx.x * blockDim.x + threadIdx.x;
  const int total = (E + N) * QP;            // max 3.3M*16 = 52.8M < 2^31
  if (gid >= total) return;
  const int e = gid / QP, q = gid % QP;
  const int t = q * 4;
  const int h = t / C;
  int src, dst;
  if (e < E) { src = ei[e]; dst = ei[E + e]; } else { src = dst = e - E; }
  const float x = leaky(als[src * H + h] + ald[dst * H + h]);
  const float a = __expf(x - m[dst * H + h]) / (denom[dst * H + h] + 1e-16f);
  const float4 f = *(const float4*)(feat + (size_t)src * HC + t);
  float* __restrict__ ap = agg + (size_t)dst * HC + t;
  atomicAdd(ap + 0, a * f.x);
  atomicAdd(ap + 1, a * f.y);
  atomicAdd(ap + 2, a * f.z);
  atomicAdd(ap + 3, a * f.w);
}

// out = elu(agg + bias), layer-1 activation.
__global__ void gat_bias_elu(const float* __restrict__ agg,
                             const float* __restrict__ b,
                             float* __restrict__ out, int N) {
  const int gid = blockIdx.x * blockDim.x + threadIdx.x;
  if (gid >= N * HC1) return;
  const float x = agg[gid] + b[gid % HC1];
  out[gid] = x > 0.f ? x : (__expf(x) - 1.0f);
}

// Final: log_softmax(agg2 + b2) over 16 classes; one thread per node.
__global__ void gat_final_lsm(const float* __restrict__ agg2,
                              const float* __restrict__ b2,
                              float* __restrict__ out, int N) {
  const int n = blockIdx.x * blockDim.x + threadIdx.x;
  if (n >= N) return;
  float v[NCLS];
  float mx = -INFINITY;
#pragma unroll
  for (int k = 0; k < NCLS; ++k) {
    v[k] = agg2[(size_t)n * NCLS + k] + b2[k];
    mx = fmaxf(mx, v[k]);
  }
  float s = 0.f;
#pragma unroll
  for (int k = 0; k < NCLS; ++k) s += __expf(v[k] - mx);
  const float lse = __logf(s);
#pragma unroll
  for (int k = 0; k < NCLS; ++k) out[(size_t)n * NCLS + k] = v[k] - mx - lse;
}

// ---------------------------------------------------------------------------
// Host launch
// ---------------------------------------------------------------------------
static inline int cdiv(long long a, int b) { return (int)((a + b - 1) / b); }

extern "C" void kernel_launch(void* const* d_in, const int* in_sizes, int n_in,
                              void* d_out, int out_size, void* d_ws, size_t ws_size,
                              hipStream_t stream) {
  const float* x     = (const float*)d_in[0];
  const int*   ei    = (const int*)  d_in[1];   // [2, E] flat: src then dst
  const float* W1    = (const float*)d_in[2];
  const float* asrc1 = (const float*)d_in[3];
  const float* adst1 = (const float*)d_in[4];
  const float* b1    = (const float*)d_in[5];
  const float* W2    = (const float*)d_in[6];
  const float* asrc2 = (const float*)d_in[7];
  const float* adst2 = (const float*)d_in[8];
  const float* b2    = (const float*)d_in[9];
  float* out = (float*)d_out;

  const int N = in_sizes[0] / F_IN;   // 100000 (divisible by 16)
  const int E = in_sizes[1] / 2;      // 3200000
  const int rowTiles = N / 16;

  // Workspace layout (floats).
  float* ws    = (float*)d_ws;
  float* H1    = ws;                       // N*64  (later reused for ELU output)
  float* als1  = H1   + (size_t)N * HC1;   // N*8
  float* ald1  = als1 + (size_t)N * HEADS; // N*8
  float* m1    = ald1 + (size_t)N * HEADS; // N*8
  float* den1  = m1   + (size_t)N * HEADS; // N*8
  float* agg1  = den1 + (size_t)N * HEADS; // N*64
  float* H2    = agg1 + (size_t)N * HC1;   // N*16
  float* als2  = H2   + (size_t)N * NCLS;  // N
  float* ald2  = als2 + (size_t)N;         // N
  float* m2    = ald2 + (size_t)N;         // N
  float* den2  = m2   + (size_t)N;         // N
  float* agg2  = den2 + (size_t)N;         // N*16

  const int TB = 256;

  // ---------------- Layer 1 ----------------
  gat_gemm1_wmma<<<cdiv(rowTiles, 8), 256, 0, stream>>>(x, W1, H1, rowTiles);
  gat_logits1<<<cdiv((long long)N * HEADS, TB), TB, 0, stream>>>(H1, asrc1, adst1, als1, ald1, N);

  fill_f32<<<cdiv((long long)N * HEADS, TB), TB, 0, stream>>>(m1, -INFINITY, N * HEADS);
  fill_f32<<<cdiv((long long)N * HEADS, TB), TB, 0, stream>>>(den1, 0.f, N * HEADS);
  fill_f32<<<cdiv((long long)N * HC1, TB), TB, 0, stream>>>(agg1, 0.f, N * HC1);

  const long long EH1 = (long long)(E + N) * HEADS;
  gat_edge_max   <<<cdiv(EH1, TB), TB, 0, stream>>>(ei, als1, ald1, m1, E, N, HEADS);
  gat_edge_expsum<<<cdiv(EH1, TB), TB, 0, stream>>>(ei, als1, ald1, m1, den1, E, N, HEADS);
  const long long EQ1 = (long long)(E + N) * (HC1 / 4);
  gat_edge_agg4  <<<cdiv(EQ1, TB), TB, 0, stream>>>(ei, als1, ald1, m1, den1, H1, agg1,
                                                    E, N, HEADS, HID);

  // ELU(agg1 + b1) -> reuse H1 buffer as layer-2 input.
  gat_bias_elu<<<cdiv((long long)N * HC1, TB), TB, 0, stream>>>(agg1, b1, H1, N);

  // ---------------- Layer 2 ----------------
  gat_gemm2_wmma<<<cdiv(rowTiles, 8), 256, 0, stream>>>(H1, W2, H2, rowTiles);
  gat_logits2<<<cdiv(N, TB), TB, 0, stream>>>(H2, asrc2, adst2, als2, ald2, N);

  fill_f32<<<cdiv(N, TB), TB, 0, stream>>>(m2, -INFINITY, N);
  fill_f32<<<cdiv(N, TB), TB, 0, stream>>>(den2, 0.f, N);
  fill_f32<<<cdiv((long long)N * NCLS, TB), TB, 0, stream>>>(agg2, 0.f, N * NCLS);

  const long long EH2 = (long long)(E + N);
  gat_edge_max   <<<cdiv(EH2, TB), TB, 0, stream>>>(ei, als2, ald2, m2, E, N, 1);
  gat_edge_expsum<<<cdiv(EH2, TB), TB, 0, stream>>>(ei, als2, ald2, m2, den2, E, N, 1);
  const long long EQ2 = (long long)(E + N) * (NCLS / 4);
  gat_edge_agg4  <<<cdiv(EQ2, TB), TB, 0, stream>>>(ei, als2, ald2, m2, den2, H2, agg2,
                                                    E, N, 1, NCLS);

  // heads=1, concat=False -> mean over 1 head is identity; add b2 + log_softmax.
  gat_final_lsm<<<cdiv(N, TB), TB, 0, stream>>>(agg2, b2, out, N);
}